// GCNLayer_317827580688
// MI455X (gfx1250) — compile-verified
//
#include <hip/hip_runtime.h>
#include <hip/hip_bf16.h>
#include <stddef.h>

#define D_FEAT 128

typedef __attribute__((ext_vector_type(2))) float v2f;
typedef __attribute__((ext_vector_type(8))) float v8f;

// ---------------------------------------------------------------------------
// Kernel 1: zero the accumulation workspace h [nNodes * 128] (float4 stores)
// ---------------------------------------------------------------------------
__global__ void __launch_bounds__(256) zero_ws_kernel(float4* __restrict__ h, int n4) {
    int i = blockIdx.x * 256 + threadIdx.x;
    if (i < n4) h[i] = make_float4(0.f, 0.f, 0.f, 0.f);
}

// ---------------------------------------------------------------------------
// Kernel 2: edge scatter.  One wave32 per edge; each lane gathers a float4
// (global_load_b128) from feature[src] and does 4 global_atomic_add_f32 into
// h[dst].  Both feature (25.6MB) and h (25.6MB) are L2-resident on MI455X.
// ---------------------------------------------------------------------------
__device__ __forceinline__ void atomic_add_f32(float* p, float v) {
#if defined(__AMDGCN__)
    unsafeAtomicAdd(p, v);   // lowers to hardware global_atomic_add_f32
#else
    atomicAdd(p, v);
#endif
}

__global__ void __launch_bounds__(256) edge_scatter_kernel(
    const float* __restrict__ feat, const int* __restrict__ src,
    const int* __restrict__ dst, float* __restrict__ h, int nEdges) {
    int tid  = blockIdx.x * 256 + threadIdx.x;
    int edge = tid >> 5;
    int lane = tid & 31;
    if (edge >= nEdges) return;
    int s = src[edge];
    int d = dst[edge];
    const float4 v = *(const float4*)(feat + (size_t)s * D_FEAT + lane * 4);
    float* hp = h + (size_t)d * D_FEAT + lane * 4;
    atomic_add_f32(hp + 0, v.x);
    atomic_add_f32(hp + 1, v.y);
    atomic_add_f32(hp + 2, v.z);
    atomic_add_f32(hp + 3, v.w);
}

// ---------------------------------------------------------------------------
// Kernel 3: out = h @ W^T + b  via V_WMMA_F32_16X16X4_F32 (exact fp32 math).
//
// One wave32 computes one 16(M=nodes) x 16(N=out-dims) tile, K=128 in steps
// of 4 -> 32 WMMA ops per tile.  Block = 256 threads = 8 waves = the 8
// N-tiles covering all 128 output dims for one group of 16 node rows.
//
// ISA VGPR layouts (CDNA5 ISA):
//   A 16x4 f32 : lane L<16 -> row M=L,   VGPR0=K0, VGPR1=K1
//                lane L>=16-> row M=L-16,VGPR0=K2, VGPR1=K3
//   B 4x16 f32 : symmetric (lane = column N, lane-half selects K pair)
//   C/D 16x16  : 8 VGPRs; lane N=L%16; VGPR r -> M = r + 8*(L/16)
// ---------------------------------------------------------------------------
__global__ void __launch_bounds__(256) gemm_bias_kernel(
    const float* __restrict__ h, const float* __restrict__ W,
    const float* __restrict__ bias, float* __restrict__ out, int nNodes) {
    const int wave  = threadIdx.x >> 5;      // 0..7 -> N tile
    const int lane  = threadIdx.x & 31;
    const int half  = lane >> 4;             // 0 or 1 (K-pair select)
    const int l16   = lane & 15;
    const int mBase = blockIdx.x * 16;       // node row base
    const int nBase = wave * 16;             // output column base

    // clamp A-row read for a possible tail tile (stores are guarded below)
    int aRow = mBase + l16;
    if (aRow >= nNodes) aRow = nNodes - 1;

    const float* __restrict__ Arow = h + (size_t)aRow * D_FEAT;           // h[m][*]
    const float* __restrict__ Brow = W + (size_t)(nBase + l16) * D_FEAT;  // W[n][*]

    v8f c = {0.f, 0.f, 0.f, 0.f, 0.f, 0.f, 0.f, 0.f};

#pragma unroll
    for (int k0 = 0; k0 < D_FEAT; k0 += 4) {
        const int ka = k0 + 2 * half;
        v2f a, b;
        a.x = Arow[ka];
        a.y = Arow[ka + 1];
        b.x = Brow[ka];
        b.y = Brow[ka + 1];
        // D = A * B + C   (v_wmma_f32_16x16x4_f32)
        c = __builtin_amdgcn_wmma_f32_16x16x4_f32(
                /*neg_a=*/false, a, /*neg_b=*/false, b,
                /*c_mod=*/(short)0, c, /*reuse_a=*/false, /*reuse_b=*/false);
    }

    const float bv = bias[nBase + l16];      // lane's output column
    float* __restrict__ outBase = out + (size_t)mBase * D_FEAT + nBase;
    const int rowOff = half * 8;
#pragma unroll
    for (int r = 0; r < 8; ++r) {
        const int m = r + rowOff;
        if (mBase + m < nNodes)
            outBase[(size_t)m * D_FEAT + l16] = c[r] + bv;
    }
}

// ---------------------------------------------------------------------------
// Host-side launcher
// ---------------------------------------------------------------------------
extern "C" void kernel_launch(void* const* d_in, const int* in_sizes, int n_in,
                              void* d_out, int out_size, void* d_ws, size_t ws_size,
                              hipStream_t stream) {
    const float* feature = (const float*)d_in[0];
    const int*   src     = (const int*)d_in[1];
    const int*   dst     = (const int*)d_in[2];
    const float* W       = (const float*)d_in[3];
    const float* b       = (const float*)d_in[4];
    float*       out     = (float*)d_out;

    const int nNodes = in_sizes[0] / D_FEAT;
    const int nEdges = in_sizes[1];

    float* h = (float*)d_ws;                 // [nNodes * 128] accumulator

    // 1) zero h
    {
        const int n4 = nNodes * (D_FEAT / 4);
        const int blocks = (n4 + 255) / 256;
        zero_ws_kernel<<<blocks, 256, 0, stream>>>((float4*)h, n4);
    }
    // 2) scatter-add edges (1 wave per edge)
    {
        const long long threads = (long long)nEdges * 32;
        const int blocks = (int)((threads + 255) / 256);
        edge_scatter_kernel<<<blocks, 256, 0, stream>>>(feature, src, dst, h, nEdges);
    }
    // 3) WMMA GEMM + bias
    {
        const int blocks = (nNodes + 15) / 16;   // one block per 16 node rows
        gemm_bias_kernel<<<blocks, 256, 0, stream>>>(h, W, b, out, nNodes);
    }
}